// GPUSkinningOperator_68186900791877
// MI455X (gfx1250) — compile-verified
//
#include <hip/hip_runtime.h>
#include <stdint.h>

#define THREADS 256
#define MAX_BONES 256   // reference uses M=256 -> 16 KB palette, fits LDS easily

// GPU skinning: out_v[n] = sum_k w[n,k] * ( (M[idx[n,k]] @ [v,1])[:3] / (M @ [v,1])[3] )
//               out_n[n] = sum_k w[n,k] * ( M[idx[n,k]][:3,:3] @ n )
// Memory-bound (~96 B/vertex, ~201 MB total -> ~8.6 us at 23.3 TB/s).
// CDNA5 path used: async global->LDS staging of the bone palette
// (global_load_async_to_lds_b128 + s_wait_asynccnt), LDS gathers via ds_load_b128.
__global__ __launch_bounds__(THREADS)
void skin_kernel(const float* __restrict__ verts,
                 const float* __restrict__ norms,
                 const float* __restrict__ wts,
                 const long long* __restrict__ bidx,
                 const float* __restrict__ mats,
                 float* __restrict__ out,
                 int nverts, int nmats) {
    __shared__ __attribute__((aligned(16))) float smem[MAX_BONES * 16];

    // ---- Stage bone palette into LDS with gfx1250 async DMA (ASYNCcnt-tracked).
    {
        // Low 32 bits of a generic pointer to LDS == LDS byte offset (flat aperture).
        const uint32_t lds_base = (uint32_t)(uintptr_t)(&smem[0]);
        const uint64_t gbase = (uint64_t)(uintptr_t)mats;
        const int chunks = nmats * 4;  // 16-byte chunks (nmats * 64 B)
        for (int c = threadIdx.x; c < chunks; c += THREADS) {
            uint32_t lds_addr = lds_base + (uint32_t)(c * 16);
            uint64_t gaddr = gbase + (uint64_t)(c * 16);
            asm volatile("global_load_async_to_lds_b128 %0, %1, off"
                         :: "v"(lds_addr), "v"(gaddr)
                         : "memory");
        }
        asm volatile("s_wait_asynccnt 0" ::: "memory");
    }
    __syncthreads();

    const int stride = gridDim.x * THREADS;
    for (int v = (int)(blockIdx.x * THREADS + threadIdx.x); v < nverts; v += stride) {
        // Coalesced per-vertex streams.
        const float vx = verts[3 * v + 0];
        const float vy = verts[3 * v + 1];
        const float vz = verts[3 * v + 2];
        const float nx0 = norms[3 * v + 0];
        const float ny0 = norms[3 * v + 1];
        const float nz0 = norms[3 * v + 2];
        const float4 w4 = ((const float4*)wts)[v];
        const float wk[4] = {w4.x, w4.y, w4.z, w4.w};
        const long long* ip = bidx + 4ll * v;

        float ov0 = 0.f, ov1 = 0.f, ov2 = 0.f;
        float on0 = 0.f, on1 = 0.f, on2 = 0.f;

#pragma unroll
        for (int k = 0; k < 4; ++k) {
            const int b = (int)ip[k];
            const float* m = &smem[b * 16];  // 64 B, 16 B aligned -> ds_load_b128 x4
            // Homogeneous transform of position.
            const float t0 = m[0] * vx + m[1] * vy + m[2] * vz + m[3];
            const float t1 = m[4] * vx + m[5] * vy + m[6] * vz + m[7];
            const float t2 = m[8] * vx + m[9] * vy + m[10] * vz + m[11];
            const float t3 = m[12] * vx + m[13] * vy + m[14] * vz + m[15];
            const float winv = wk[k] / t3;  // perspective divide folded into weight
            ov0 += winv * t0;
            ov1 += winv * t1;
            ov2 += winv * t2;
            // Rotation-only transform of normal.
            const float r0 = m[0] * nx0 + m[1] * ny0 + m[2] * nz0;
            const float r1 = m[4] * nx0 + m[5] * ny0 + m[6] * nz0;
            const float r2 = m[8] * nx0 + m[9] * ny0 + m[10] * nz0;
            on0 += wk[k] * r0;
            on1 += wk[k] * r1;
            on2 += wk[k] * r2;
        }

        float* po = out + 3ll * v;  // out_v: first nverts*3 floats
        po[0] = ov0;
        po[1] = ov1;
        po[2] = ov2;
        float* pn = out + 3ll * nverts + 3ll * v;  // out_n: next nverts*3 floats
        pn[0] = on0;
        pn[1] = on1;
        pn[2] = on2;
    }
}

extern "C" void kernel_launch(void* const* d_in, const int* in_sizes, int n_in,
                              void* d_out, int out_size, void* d_ws, size_t ws_size,
                              hipStream_t stream) {
    (void)n_in;
    (void)d_ws;
    (void)ws_size;
    (void)out_size;
    const float* verts = (const float*)d_in[0];        // (N,3) f32
    const float* norms = (const float*)d_in[1];        // (N,3) f32
    const float* wts = (const float*)d_in[2];          // (N,4) f32
    const long long* bidx = (const long long*)d_in[3]; // (N,4) i64
    const float* mats = (const float*)d_in[4];         // (M,4,4) f32
    float* out = (float*)d_out;                        // (N,3) ++ (N,3) f32

    const int nverts = in_sizes[0] / 3;
    const int nmats = in_sizes[4] / 16;

    int blocks = (nverts + THREADS - 1) / THREADS;
    if (blocks > 4096) blocks = 4096;  // grid-stride; amortize palette staging
    if (blocks < 1) blocks = 1;

    skin_kernel<<<blocks, THREADS, 0, stream>>>(verts, norms, wts, bidx, mats,
                                                out, nverts, nmats);
}